// PGBM_38740605010080
// MI455X (gfx1250) — compile-verified
//
#include <hip/hip_runtime.h>

#define NFEAT 32
#define NBIN  256
#define HIST_N (2 * NFEAT * NBIN)   // 16384 floats = 64 KB, layout [v][f][b]

typedef __attribute__((ext_vector_type(4))) int   v4i;
typedef __attribute__((ext_vector_type(2))) float v2f;
typedef __attribute__((ext_vector_type(8))) float v8f;

// ---------------------------------------------------------------------------
// Stage 1: per-workgroup LDS histogram of gradient & hessian over 32 features
// x 256 bins, flushed with global f32 atomics into acc[16384] (d_ws).
// Address (v*8192 + f*256 + b): bank = b mod 64 -> full 64-bank spread per op.
// ---------------------------------------------------------------------------
__global__ __launch_bounds__(512) void pgbm_hist_kernel(
    const int* __restrict__ X, const float* __restrict__ g,
    const float* __restrict__ h, float* __restrict__ acc, int N)
{
    __shared__ float sh[HIST_N];
    for (int i = threadIdx.x; i < HIST_N; i += blockDim.x) sh[i] = 0.0f;
    __syncthreads();

    const int tid    = blockIdx.x * blockDim.x + threadIdx.x;
    const int stride = gridDim.x * blockDim.x;

    for (int i = tid; i < N; i += stride) {
        // Prefetch next grid-stride row (speculative; OOB is dropped by HW).
        __builtin_prefetch(X + (size_t)(i + stride) * NFEAT, 0, 1);

        const float gv = __builtin_nontemporal_load(g + i);
        const float hv = __builtin_nontemporal_load(h + i);
        const v4i* __restrict__ row = (const v4i*)(X + (size_t)i * NFEAT);
#pragma unroll
        for (int c = 0; c < 8; ++c) {
            v4i b4 = __builtin_nontemporal_load(row + c);   // 128-bit load
            const int f0 = c * 4;
            const int i0 = (f0 + 0) * NBIN + b4.x;
            const int i1 = (f0 + 1) * NBIN + b4.y;
            const int i2 = (f0 + 2) * NBIN + b4.z;
            const int i3 = (f0 + 3) * NBIN + b4.w;
            atomicAdd(&sh[i0], gv);  atomicAdd(&sh[i0 + NFEAT * NBIN], hv);
            atomicAdd(&sh[i1], gv);  atomicAdd(&sh[i1 + NFEAT * NBIN], hv);
            atomicAdd(&sh[i2], gv);  atomicAdd(&sh[i2 + NFEAT * NBIN], hv);
            atomicAdd(&sh[i3], gv);  atomicAdd(&sh[i3 + NFEAT * NBIN], hv);
        }
    }
    __syncthreads();
    for (int i = threadIdx.x; i < HIST_N; i += blockDim.x)
        atomicAdd(&acc[i], sh[i]);          // global_atomic_add_f32
}

// ---------------------------------------------------------------------------
// Stage 2: out[m, n] = sum_{k > n} hist[m, k]  (m = value*32 + feature)
// == Hist(64x256) x StrictLowerOnes(256x256), done with V_WMMA_F32_16X16X4_F32.
// One wave per 16x16 output tile; B (0/1 triangle) built in registers.
// ---------------------------------------------------------------------------
__global__ __launch_bounds__(32) void pgbm_finalize_kernel(
    const float* __restrict__ acc, float* __restrict__ out)
{
    const int mt   = blockIdx.x >> 4;       // 0..3  : 16-row tile of 64 rows
    const int nt   = blockIdx.x & 15;       // 0..15 : 16-bin output tile
    const int lane = threadIdx.x;           // full wave32, EXEC all ones
    const int l15  = lane & 15;
    const int half = lane >> 4;             // selects K pair (ISA A/B layout)
    const int mrow = (mt << 4) + l15;       // A-matrix row for this lane
    const int ncol = (nt << 4) + l15;       // output column for this lane

    v8f c = {};
    for (int k0 = 0; k0 < NBIN; k0 += 4) {
        const int k = k0 + (half << 1);
        v2f a, b;
        a.x = acc[mrow * NBIN + k];         // A[m, k]   (16x4 f32 fragment)
        a.y = acc[mrow * NBIN + k + 1];     // A[m, k+1]
        b.x = (k     > ncol) ? 1.0f : 0.0f; // B[k,   n] strict lower triangle
        b.y = (k + 1 > ncol) ? 1.0f : 0.0f; // B[k+1, n]
        c = __builtin_amdgcn_wmma_f32_16x16x4_f32(
                /*neg_a=*/false, a, /*neg_b=*/false, b,
                /*c_mod=*/(short)0, c, /*reuse_a=*/false, /*reuse_b=*/false);
    }
#pragma unroll
    for (int r = 0; r < 8; ++r) {           // D layout: VGPR r -> row r (+8)
        const int row = (mt << 4) + r + (half << 3);
        out[row * NBIN + ncol] = c[r];
    }
}

// ---------------------------------------------------------------------------
extern "C" void kernel_launch(void* const* d_in, const int* in_sizes, int n_in,
                              void* d_out, int out_size, void* d_ws, size_t ws_size,
                              hipStream_t stream) {
    const int*   X = (const int*)d_in[0];    // [N, 32] int32 bins
    const float* g = (const float*)d_in[1];  // [N] gradient
    const float* h = (const float*)d_in[2];  // [N] hessian
    float* out = (float*)d_out;              // [2, 32, 256] (Gl then Hl)
    float* acc = (float*)d_ws;               // 64 KB accumulator
    const int N = in_sizes[1];

    hipMemsetAsync(acc, 0, HIST_N * sizeof(float), stream);  // capturable
    pgbm_hist_kernel<<<512, 512, 0, stream>>>(X, g, h, acc, N);
    pgbm_finalize_kernel<<<64, 32, 0, stream>>>(acc, out);
}